// GCN_1357209665582
// MI455X (gfx1250) — compile-verified
//
#include <hip/hip_runtime.h>

#define N_TOTAL    64000
#define NODE_COUNT 1000
#define BATCH      64
#define STATE      16
#define HID        32
#define N_EDGES    2048000
#define E_TOTAL    (N_EDGES + N_TOTAL)   // edges + self loops

typedef float v2f __attribute__((ext_vector_type(2)));
typedef float v8f __attribute__((ext_vector_type(8)));

// explicit global-address-space pointers: force global_load/global_store
// lowering (and pure LOADcnt/STOREcnt tracking) instead of flat_*.
typedef const __attribute__((address_space(1))) float* gcfp;
typedef __attribute__((address_space(1))) float*       gfp;

// ---------------------------------------------------------------- utilities
__global__ void k_zero(float* __restrict__ p, int n) {
  int i = blockIdx.x * blockDim.x + threadIdx.x;
  if (i < n) p[i] = 0.0f;
}

// deg[c] += 1 for every real edge target (self-loops folded in later as +1)
__global__ void k_degree(const int* __restrict__ cols, float* __restrict__ deg) {
  int e = blockIdx.x * blockDim.x + threadIdx.x;
  if (e < N_EDGES) atomicAdd(&deg[cols[e]], 1.0f);
}

// in-place deg -> dinv ; self-loop contributes +1, so deg+1 > 0 always
__global__ void k_dinv(float* __restrict__ deg) {
  int i = blockIdx.x * blockDim.x + threadIdx.x;
  if (i < N_TOTAL) deg[i] = rsqrtf(deg[i] + 1.0f);
}

// norm[e] = dinv[row]*dinv[col]; tail e in [N_EDGES, E_TOTAL) are self loops
__global__ void k_norm(const int* __restrict__ rows, const int* __restrict__ cols,
                       const float* __restrict__ dinv, float* __restrict__ norm) {
  int e = blockIdx.x * blockDim.x + threadIdx.x;
  if (e >= E_TOTAL) return;
  int r, c;
  if (e < N_EDGES) { r = rows[e]; c = cols[e]; }
  else             { r = c = e - N_EDGES; }
  norm[e] = dinv[r] * dinv[c];
}

// one wave32 per edge, one lane per feature (HID == 32 == wave32)
__global__ void k_scatter(const int* __restrict__ rows, const int* __restrict__ cols,
                          const float* __restrict__ norm,
                          const float* __restrict__ h, float* __restrict__ out) {
  int t = blockIdx.x * blockDim.x + threadIdx.x;
  int e = t >> 5;          // edge handled by this wave
  int j = t & 31;          // feature lane
  if (e >= E_TOTAL) return;
  int r, c;
  if (e < N_EDGES) { r = rows[e]; c = cols[e]; }
  else             { r = c = e - N_EDGES; }
  float w = norm[e];       // wave-uniform
  gcfp hg = (gcfp)h;
  atomicAdd(&out[(unsigned)c * HID + j], hg[(unsigned)r * HID + j] * w);
}

__global__ void k_bias_relu(float* __restrict__ x, const float* __restrict__ b, int n) {
  int i = blockIdx.x * blockDim.x + threadIdx.x;
  if (i < n) {
    float v = x[i] + b[i & (HID - 1)];
    x[i] = v > 0.0f ? v : 0.0f;
  }
}

// ------------------------------------------------- fp32 WMMA GEMM (16x16x4)
// C[M,N] = A[M,K] @ B[K,N] (+bias[n]) (+relu). Row-major, lda=K, ldb=ldc=N.
// blockDim = 128 (4 waves); wave w computes M-tile (blockIdx.y*4 + w) and
// NT consecutive 16-wide N-tiles (one A fragment feeds NT WMMAs).
// All global accesses go through address_space(1) pointers with 32-bit
// loop-carried offsets -> saddr-form global_load, no 64-bit VALU adds.
// RAGGED: clamped-address loads + value select (no EXEC masking), stores
// predicated per lane only.
template <int NT, bool RAGGED, bool BIAS, bool RELU>
__global__ void k_gemm_wmma(const float* __restrict__ A, const float* __restrict__ B,
                            const float* __restrict__ bias, float* __restrict__ C,
                            int M, int N, int K) {
  int wave  = threadIdx.x >> 5;
  int lane  = threadIdx.x & 31;
  int mtile = blockIdx.y * 4 + wave;
  int m0    = mtile * 16;
  if (m0 >= M) return;                     // wave-uniform exit
  int n0   = blockIdx.x * (16 * NT);
  int half = lane >> 4;                    // 0: lanes 0-15, 1: lanes 16-31
  int lm   = lane & 15;

  gcfp Ag = (gcfp)A;
  gcfp Bg = (gcfp)B;

  int  ncol[NT];                           // output column per N-tile
  bool nok[NT];
  unsigned boff[NT];                       // B load offset (clamped in bounds)
#pragma unroll
  for (int t = 0; t < NT; ++t) {
    ncol[t] = n0 + t * 16 + lm;
    int nld;
    if (RAGGED) {
      nok[t] = ncol[t] < N;
      nld    = nok[t] ? ncol[t] : (N - 1); // in-bounds address, value zeroed below
    } else {
      nok[t] = true;
      nld    = ncol[t];
    }
    boff[t] = (unsigned)(half * 2) * (unsigned)N + (unsigned)nld;
  }

  // A: lane lm holds row m0+lm; VGPR v holds K = k0 + half*2 + v (contig pair)
  unsigned aoff = (unsigned)(m0 + lm) * (unsigned)K + (unsigned)(half * 2);

  v8f acc[NT];
#pragma unroll
  for (int t = 0; t < NT; ++t) acc[t] = (v8f){0.f,0.f,0.f,0.f,0.f,0.f,0.f,0.f};

  for (int k0 = 0; k0 < K; k0 += 4) {
    v2f a;
    a.x = Ag[aoff];
    a.y = Ag[aoff + 1];
#pragma unroll
    for (int t = 0; t < NT; ++t) {
      float b0 = Bg[boff[t]];
      float b1 = Bg[boff[t] + (unsigned)N];
      if (RAGGED) { b0 = nok[t] ? b0 : 0.0f; b1 = nok[t] ? b1 : 0.0f; }
      v2f b; b.x = b0; b.y = b1;
      acc[t] = __builtin_amdgcn_wmma_f32_16x16x4_f32(
          /*neg_a=*/false, a, /*neg_b=*/false, b,
          /*c_mod=*/(short)0, acc[t], /*reuse_a=*/false, /*reuse_b=*/false);
      boff[t] += 4u * (unsigned)N;
    }
    aoff += 4u;
  }

  gfp Cg = (gfp)C;
  gcfp biasg = (gcfp)bias;
#pragma unroll
  for (int t = 0; t < NT; ++t) {
    float bv = BIAS ? biasg[nok[t] ? ncol[t] : 0] : 0.0f;
    if (!RAGGED || nok[t]) {
#pragma unroll
      for (int r = 0; r < 8; ++r) {        // VGPR r: M = r (+8 for upper half)
        int m = m0 + r + half * 8;
        float v = acc[t][r] + bv;
        if (RELU) v = v > 0.0f ? v : 0.0f;
        Cg[(unsigned)m * (unsigned)N + (unsigned)ncol[t]] = v;
      }
    }
  }
}

// value head: V[b] = hidv[b,:]·Wv2 + bv2   (64 dots of length 256)
__global__ void k_value(const float* __restrict__ hidv, const float* __restrict__ Wv2,
                        const float* __restrict__ bv2, float* __restrict__ out) {
  int b = blockIdx.x * blockDim.x + threadIdx.x;
  if (b < BATCH) {
    float s = bv2[0];
    for (int k = 0; k < 256; ++k) s += hidv[b * 256 + k] * Wv2[k];
    out[b] = s;
  }
}

// ------------------------------------------------------------------ launcher
static inline int cdiv(long long a, int b) { return (int)((a + b - 1) / b); }

extern "C" void kernel_launch(void* const* d_in, const int* in_sizes, int n_in,
                              void* d_out, int out_size, void* d_ws, size_t ws_size,
                              hipStream_t stream) {
  (void)in_sizes; (void)n_in; (void)out_size; (void)ws_size;

  const float* x    = (const float*)d_in[0];
  const int*   ei   = (const int*)d_in[1];          // [2, N_EDGES] int32
  const int*   rows = ei;
  const int*   cols = ei + N_EDGES;
  const float* W[5]  = {(const float*)d_in[2], (const float*)d_in[4], (const float*)d_in[6],
                        (const float*)d_in[8], (const float*)d_in[10]};
  const float* bb[5] = {(const float*)d_in[3], (const float*)d_in[5], (const float*)d_in[7],
                        (const float*)d_in[9], (const float*)d_in[11]};
  const float* Wp1 = (const float*)d_in[12]; const float* bp1 = (const float*)d_in[13];
  const float* Wp2 = (const float*)d_in[14]; const float* bp2 = (const float*)d_in[15];
  const float* Wv1 = (const float*)d_in[16]; const float* bv1 = (const float*)d_in[17];
  const float* Wv2 = (const float*)d_in[18]; const float* bv2 = (const float*)d_in[19];
  float* out = (float*)d_out;

  // workspace carving (~25.3 MB of f32)
  float* ws   = (float*)d_ws;
  float* deg  = ws;                            // N_TOTAL   (becomes dinv in place)
  float* norm = deg  + N_TOTAL;                // E_TOTAL
  float* bufA = norm + E_TOTAL;                // N_TOTAL*HID  (GEMM out / gather src)
  float* bufB = bufA + (size_t)N_TOTAL * HID;  // N_TOTAL*HID  (aggregate / layer out)
  float* hidp = bufB + (size_t)N_TOTAL * HID;  // BATCH*256
  float* hidv = hidp + BATCH * 256;            // BATCH*256

  const int TB = 256;

  // ---- normalization coefficients (shared across all 5 layers)
  k_zero  <<<cdiv(N_TOTAL, TB), TB, 0, stream>>>(deg, N_TOTAL);
  k_degree<<<cdiv(N_EDGES, TB), TB, 0, stream>>>(cols, deg);
  k_dinv  <<<cdiv(N_TOTAL, TB), TB, 0, stream>>>(deg);
  k_norm  <<<cdiv(E_TOTAL, TB), TB, 0, stream>>>(rows, cols, deg, norm);

  // ---- 5 GCN layers: GEMM -> zero -> scatter -> bias+relu
  const float* in = x;
  int Kdim = STATE;
  dim3 layer_grid(1, N_TOTAL / 64);            // N=32 -> one NT=2 tile; 1000 M-groups
  int scat_blocks = cdiv((long long)E_TOTAL * 32, TB);
  for (int l = 0; l < 5; ++l) {
    k_gemm_wmma<2, false, false, false><<<layer_grid, 128, 0, stream>>>(
        in, W[l], nullptr, bufA, N_TOTAL, HID, Kdim);
    k_zero     <<<cdiv((long long)N_TOTAL * HID, TB), TB, 0, stream>>>(bufB, N_TOTAL * HID);
    k_scatter  <<<scat_blocks, TB, 0, stream>>>(rows, cols, norm, bufA, bufB);
    k_bias_relu<<<cdiv((long long)N_TOTAL * HID, TB), TB, 0, stream>>>(bufB, bb[l],
                                                                       N_TOTAL * HID);
    in = bufB;
    Kdim = HID;
  }

  // ---- heads: flat = bufB viewed as [BATCH, NODE_COUNT*HID] (row-major contiguous)
  dim3 g1(256 / 32, cdiv(BATCH, 64));          // NT=2 -> 8 N-groups; 4 M-tiles (1 group)
  k_gemm_wmma<2, false, true, true><<<g1, 128, 0, stream>>>(
      bufB, Wp1, bp1, hidp, BATCH, 256, NODE_COUNT * HID);
  k_gemm_wmma<2, false, true, true><<<g1, 128, 0, stream>>>(
      bufB, Wv1, bv1, hidv, BATCH, 256, NODE_COUNT * HID);

  dim3 g2(cdiv(NODE_COUNT, 16), cdiv(BATCH, 64));  // ragged N=1000 (62 full + 1 partial)
  k_gemm_wmma<1, true, true, false><<<g2, 128, 0, stream>>>(
      hidp, Wp2, bp2, out, BATCH, NODE_COUNT, 256);

  k_value<<<1, 64, 0, stream>>>(hidv, Wv2, bv2, out + (size_t)BATCH * NODE_COUNT);
}